// LMMSETISTA_54674933678597
// MI455X (gfx1250) — compile-verified
//
#include <hip/hip_runtime.h>

#define LDS_LD 66      // padded row stride (floats) to avoid LDS bank conflicts
#define NMAT 64
#define EPSV 1e-9f
#define NS_ITERS 8     // Newton-Schulz steps for SPD inverse

typedef __attribute__((ext_vector_type(2))) float v2f;
typedef __attribute__((ext_vector_type(8))) float v8f;

// Block-wide sum: wave32 shuffle reduce, then 8 wave partials in LDS.
__device__ __forceinline__ float blockReduceSum(float v, volatile float* red) {
#pragma unroll
  for (int o = 16; o > 0; o >>= 1) v += __shfl_down(v, o, 32);
  __syncthreads();                       // protect red reuse from prior call
  if ((threadIdx.x & 31) == 0) red[threadIdx.x >> 5] = v;
  __syncthreads();
  float s = red[0];
#pragma unroll
  for (int i = 1; i < 8; ++i) s += red[i];
  return s;                              // uniform across block
}

__device__ __forceinline__ float blockReduceMax(float v, volatile float* red) {
#pragma unroll
  for (int o = 16; o > 0; o >>= 1) v = fmaxf(v, __shfl_down(v, o, 32));
  __syncthreads();
  if ((threadIdx.x & 31) == 0) red[threadIdx.x >> 5] = v;
  __syncthreads();
  float s = red[0];
#pragma unroll
  for (int i = 1; i < 8; ++i) s = fmaxf(s, red[i]);
  return s;
}

// D(64x64) = alpha * op(A)(64x64) * op(B)(64x64), all in LDS.
// 16 output tiles of 16x16; 8 wave32s, 2 tiles per wave.
// Per tile: 16 k-steps of V_WMMA_F32_16X16X4_F32.
// Caller must __syncthreads() before (srcs ready) and after (dst ready).
__device__ __forceinline__ void gemm64(const float* A, bool tA,
                                       const float* B, bool tB,
                                       float* D, float alpha) {
  const int wave = threadIdx.x >> 5;
  const int lane = threadIdx.x & 31;
  const int half = lane >> 4;    // 0: lanes 0-15, 1: lanes 16-31
  const int l16  = lane & 15;
#pragma unroll
  for (int tt = 0; tt < 2; ++tt) {
    const int tile = wave + 8 * tt;      // 0..15
    const int mBase = (tile >> 2) * 16;
    const int nBase = (tile & 3) * 16;
    v8f acc = {};
#pragma unroll
    for (int k0 = 0; k0 < NMAT; k0 += 4) {
      // A fragment 16x4: lane holds row (l16), VGPR v holds k = k0+2*half+v
      const int m  = mBase + l16;
      const int ka = k0 + 2 * half;
      v2f a, b;
      if (!tA) { a.x = A[m * LDS_LD + ka];     a.y = A[m * LDS_LD + ka + 1]; }
      else     { a.x = A[ka * LDS_LD + m];     a.y = A[(ka + 1) * LDS_LD + m]; }
      // B fragment 4x16: lane holds col (l16), VGPR v holds k = k0+2*half+v
      const int n  = nBase + l16;
      if (!tB) { b.x = B[ka * LDS_LD + n];     b.y = B[(ka + 1) * LDS_LD + n]; }
      else     { b.x = B[n * LDS_LD + ka];     b.y = B[n * LDS_LD + ka + 1]; }
      acc = __builtin_amdgcn_wmma_f32_16x16x4_f32(
          /*neg_a=*/false, a, /*neg_b=*/false, b,
          /*c_mod=*/(short)0, acc, /*reuse_a=*/false, /*reuse_b=*/false);
    }
#pragma unroll
    for (int v = 0; v < 8; ++v) {
      const int m = mBase + v + 8 * half;  // C/D layout: VGPR v -> row v (+8 for hi half)
      const int n = nBase + l16;
      D[m * LDS_LD + n] = alpha * acc[v];
    }
  }
}

__global__ __launch_bounds__(256)
void oamp_lmmse_wmma_kernel(const float* __restrict__ Hg,
                            const float* __restrict__ yg,
                            const float* __restrict__ s2g,
                            const float* __restrict__ stepg,
                            float* __restrict__ out, int nstep) {
  __shared__ float sH[NMAT * LDS_LD];   // channel matrix
  __shared__ float sA[NMAT * LDS_LD];   // A = v2 HH^T + s2 I
  __shared__ float sX[NMAT * LDS_LD];   // Newton-Schulz inverse iterate
  __shared__ float sT[NMAT * LDS_LD];   // temp / G = A^{-1}H
  __shared__ float sW[NMAT * LDS_LD];   // NS pong / Wh = v2 G^T H
  __shared__ float sy[NMAT], sx[NMAT], svec[NMAT], sr[NMAT];
  __shared__ float sred[8];

  const int tid = threadIdx.x;
  const int b = blockIdx.x;
  const float* gH = Hg + (size_t)b * NMAT * NMAT;
  const float sigma2 = s2g[b];

  // Stage H into LDS (one HBM read per batch; rest of the problem lives in LDS)
  for (int idx = tid; idx < NMAT * NMAT; idx += 256) {
    const int r = idx >> 6, c = idx & 63;
    sH[r * LDS_LD + c] = gH[idx];
  }
  if (tid < NMAT) { sy[tid] = yg[(size_t)b * NMAT + tid]; sx[tid] = 0.0f; }
  __syncthreads();

  // den = tr(H^T H), constant over iterations
  float p = 0.0f;
  for (int idx = tid; idx < NMAT * NMAT; idx += 256) {
    const int r = idx >> 6, c = idx & 63;
    const float h = sH[r * LDS_LD + c];
    p += h * h;
  }
  const float den = blockReduceSum(p, sred);

  // v2 with x=0: resid = y
  p = (tid < NMAT) ? sy[tid] * sy[tid] : 0.0f;
  const float sumy2 = blockReduceSum(p, sred);
  float v2 = fmaxf((sumy2 - NMAT * sigma2) / den, EPSV);

  for (int t = 0; t < nstep; ++t) {
    const float st = stepg[t];

    // A = v2 * H H^T + sigma2 * I
    __syncthreads();
    gemm64(sH, false, sH, true, sA, v2);
    __syncthreads();
    if (tid < NMAT) sA[tid * LDS_LD + tid] += sigma2;
    __syncthreads();

    // Newton-Schulz init: X0 = A / (||A||_1 ||A||_inf)  (A symmetric)
    float rs = 0.0f;
    if (tid < NMAT)
      for (int j = 0; j < NMAT; ++j) rs += fabsf(sA[tid * LDS_LD + j]);
    const float rmax = blockReduceMax(rs, sred);
    const float c0 = 1.0f / (rmax * rmax);
    for (int idx = tid; idx < NMAT * NMAT; idx += 256) {
      const int r = idx >> 6, c = idx & 63;
      sX[r * LDS_LD + c] = c0 * sA[r * LDS_LD + c];
    }
    __syncthreads();

    // X <- X (2I - A X), ping-pong sX <-> sW; NS_ITERS even => result in sX
    float* xc = sX;
    float* xn = sW;
    for (int k = 0; k < NS_ITERS; ++k) {
      gemm64(sA, false, xc, false, sT, 1.0f);
      __syncthreads();
      for (int idx = tid; idx < NMAT * NMAT; idx += 256) {
        const int r = idx >> 6, c = idx & 63;
        sT[r * LDS_LD + c] = ((r == c) ? 2.0f : 0.0f) - sT[r * LDS_LD + c];
      }
      __syncthreads();
      gemm64(xc, false, sT, false, xn, 1.0f);
      __syncthreads();
      float* tmp = xc; xc = xn; xn = tmp;
    }

    // G = A^{-1} H  -> sT ;  Wh = v2 * G^T H -> sW
    gemm64(sX, false, sH, false, sT, 1.0f);
    __syncthreads();
    gemm64(sT, true, sH, false, sW, v2);
    __syncthreads();

    // scale = N / tr(Wh)
    p = (tid < NMAT) ? sW[tid * LDS_LD + tid] : 0.0f;
    const float tr = blockReduceSum(p, sred);
    const float scale = (float)NMAT / tr;
    const float sv = scale * v2;          // W = sv * G^T

    // resid = y - H x
    if (tid < NMAT) {
      float acc = 0.0f;
      for (int j = 0; j < NMAT; ++j) acc += sH[tid * LDS_LD + j] * sx[j];
      svec[tid] = sy[tid] - acc;
    }
    __syncthreads();
    // r = x + st * W resid = x + st*sv*(G^T resid)
    if (tid < NMAT) {
      float acc = 0.0f;
      for (int k = 0; k < NMAT; ++k) acc += sT[k * LDS_LD + tid] * svec[k];
      sr[tid] = sx[tid] + st * sv * acc;
    }
    // sum(W*W) = sv^2 * sum(G*G)
    p = 0.0f;
    for (int idx = tid; idx < NMAT * NMAT; idx += 256) {
      const int r = idx >> 6, c = idx & 63;
      const float g = sT[r * LDS_LD + c];
      p += g * g;
    }
    const float sumW2 = sv * sv * blockReduceSum(p, sred);
    // sum(C*C), C = I - st * scale * Wh
    p = 0.0f;
    for (int idx = tid; idx < NMAT * NMAT; idx += 256) {
      const int r = idx >> 6, c = idx & 63;
      const float cv = ((r == c) ? 1.0f : 0.0f) - st * scale * sW[r * LDS_LD + c];
      p += cv * cv;
    }
    const float sumC2 = blockReduceSum(p, sred);
    const float tau2 =
        fmaxf((sumC2 * v2 + st * st * sumW2 * sigma2) / (float)NMAT, EPSV);

    __syncthreads();
    if (tid < NMAT) sx[tid] = tanhf(sr[tid] / tau2);   // BPSK MMSE denoiser
    __syncthreads();

    // v2 = max((||y - H x||^2 - M s2) / den, EPS)
    p = 0.0f;
    if (tid < NMAT) {
      float acc = 0.0f;
      for (int j = 0; j < NMAT; ++j) acc += sH[tid * LDS_LD + j] * sx[j];
      const float rr = sy[tid] - acc;
      p = rr * rr;
    }
    const float sumr2 = blockReduceSum(p, sred);
    v2 = fmaxf((sumr2 - NMAT * sigma2) / den, EPSV);
  }

  __syncthreads();
  if (tid < NMAT) out[(size_t)b * NMAT + tid] = sx[tid];
}

extern "C" void kernel_launch(void* const* d_in, const int* in_sizes, int n_in,
                              void* d_out, int out_size, void* d_ws, size_t ws_size,
                              hipStream_t stream) {
  const float* H     = (const float*)d_in[0];   // [B,64,64]
  const float* y     = (const float*)d_in[1];   // [B,64,1]
  const float* s2    = (const float*)d_in[2];   // [B,1,1]
  const float* stepv = (const float*)d_in[3];   // [10]
  float* out = (float*)d_out;                   // [B,64,1]
  const int B = in_sizes[2];
  const int nstep = in_sizes[3];
  hipLaunchKernelGGL(oamp_lmmse_wmma_kernel, dim3(B), dim3(256), 0, stream,
                     H, y, s2, stepv, out, nstep);
}